// SphArr_19035295055893
// MI455X (gfx1250) — compile-verified
//
#include <hip/hip_runtime.h>

typedef float v4f __attribute__((ext_vector_type(4)));
typedef int   v4i __attribute__((vector_size(16)));   // matches builtin param pointee

#define PTS_PER_BLOCK 256

#if defined(__gfx1250__) && __has_builtin(__builtin_amdgcn_global_load_async_to_lds_b128) && __has_builtin(__builtin_amdgcn_s_wait_asynccnt)
#define HAVE_GFX1250_ASYNC 1
#else
#define HAVE_GFX1250_ASYNC 0
#endif

#if HAVE_GFX1250_ASYNC
typedef __attribute__((address_space(1))) v4i*         gv4i_p;  // global ptr to int4
typedef __attribute__((address_space(3))) v4i*         lv4i_p;  // LDS ptr to int4
typedef const __attribute__((address_space(3))) float* lf32_p;  // LDS ptr to float
#endif

__device__ __forceinline__ float fast_rsqrt(float x) {
#if __has_builtin(__builtin_amdgcn_rsqf)
    return __builtin_amdgcn_rsqf(x);   // v_rsq_f32
#else
    return rsqrtf(x);
#endif
}

// Output layout per point: 16 components (l,m) with l=0..3, m=-l..l, each (re, im).
// Derivation: Y_l^m = n_lm * P_l^|m|(ct) * e^{i|m|phi}, P_l^m = st^m * Q_l^m(ct),
// and st^m e^{im phi} = ((x+iy)/r)^m. Condon-Shortley kept (matches reference
// recurrence). For m<0: (re,im) -> ((-1)^m re, -(-1)^m im).
__global__ __launch_bounds__(PTS_PER_BLOCK) void spharm_all_kernel(
        const float* __restrict__ X, float* __restrict__ out, int npts) {
    const int lt   = threadIdx.x;
    const int base = blockIdx.x * PTS_PER_BLOCK;
    const int p    = base + lt;

    float x, y, z;

#if HAVE_GFX1250_ASYNC
    __shared__ float smem[PTS_PER_BLOCK * 3];
    const bool full = (base + PTS_PER_BLOCK) <= npts;
    if (full) {
        // Stage 256 points (3072 B) into LDS: lanes 0..191 each move 16 B.
        if (lt < (PTS_PER_BLOCK * 3) / 4) {
            const float* gsrc = X + (size_t)base * 3 + (size_t)lt * 4;
            __builtin_amdgcn_global_load_async_to_lds_b128(
                (gv4i_p)gsrc, (lv4i_p)&smem[lt * 4], 0, 0);
        }
        __builtin_amdgcn_s_wait_asynccnt(0);   // wait own wave's async ops
        __syncthreads();                        // cross-wave visibility
        // Explicit AS3 loads -> ds_load (prevents flat-load merge with tail path).
        lf32_p sm = (lf32_p)&smem[0];
        x = sm[lt * 3 + 0];
        y = sm[lt * 3 + 1];
        z = sm[lt * 3 + 2];
    } else {
        if (p >= npts) return;
        const float* g = X + (size_t)p * 3;
        x = g[0]; y = g[1]; z = g[2];
    }
#else
    if (p >= npts) return;
    const float* g = X + (size_t)p * 3;
    x = g[0]; y = g[1]; z = g[2];
#endif
    if (p >= npts) return;

    const float r2 = __builtin_fmaf(x, x, __builtin_fmaf(y, y, z * z));
    const float ir = fast_rsqrt(r2);
    const float ct = z * ir;            // cos(theta)
    const float u  = x * ir;            // st*cos(phi)
    const float v  = y * ir;            // st*sin(phi)
    const float ct2 = ct * ct;

    // (u + i v)^m  ==  st^m * e^{i m phi}
    const float c2 = u * u - v * v;
    const float s2 = 2.0f * u * v;
    const float c3 = c2 * u - s2 * v;
    const float s3 = c2 * v + s2 * u;

    // Normalizations folded with reduced Legendre Q_l^m(ct):
    const float n00 = 0.28209479177387814f;             // sqrt(1/4pi)
    const float n10 = 0.48860251190291992f;             // sqrt(3/4pi)
    const float n11 = 0.34549414947133547f;             // sqrt(3/8pi)
    const float b20 = 0.31539156525252005f * (3.0f * ct2 - 1.0f);      // n20*Q20
    const float b21 = -0.77254840404637906f * ct;                       // n21*Q21
    const float b22 = 0.38627420202318953f;                             // n22*Q22
    const float b30 = 0.37317633259011546f * ct * (5.0f * ct2 - 3.0f); // n30*Q30
    const float b31 = -0.32318018411415065f * (5.0f * ct2 - 1.0f);     // n31*Q31
    const float b32 = 1.02198547643328236f * ct;                       // n32*Q32
    const float b33 = -0.41722382363278420f;                           // n33*Q33

    // Pack 32 floats -> 8 x b128 streaming (non-temporal) stores.
    float* o = out + (size_t)p * 32;
    const v4f t0 = { n00,        0.0f,      n11 * u,   -n11 * v };  // (0,0) (1,-1)
    const v4f t1 = { n10 * ct,   0.0f,     -n11 * u,   -n11 * v };  // (1,0) (1,1)
    const v4f t2 = { b22 * c2,  -b22 * s2, -b21 * u,    b21 * v };  // (2,-2)(2,-1)
    const v4f t3 = { b20,        0.0f,      b21 * u,    b21 * v };  // (2,0) (2,1)
    const v4f t4 = { b22 * c2,   b22 * s2, -b33 * c3,   b33 * s3 }; // (2,2) (3,-3)
    const v4f t5 = { b32 * c2,  -b32 * s2, -b31 * u,    b31 * v };  // (3,-2)(3,-1)
    const v4f t6 = { b30,        0.0f,      b31 * u,    b31 * v };  // (3,0) (3,1)
    const v4f t7 = { b32 * c2,   b32 * s2,  b33 * c3,   b33 * s3 }; // (3,2) (3,3)

    __builtin_nontemporal_store(t0, (v4f*)(o + 0));
    __builtin_nontemporal_store(t1, (v4f*)(o + 4));
    __builtin_nontemporal_store(t2, (v4f*)(o + 8));
    __builtin_nontemporal_store(t3, (v4f*)(o + 12));
    __builtin_nontemporal_store(t4, (v4f*)(o + 16));
    __builtin_nontemporal_store(t5, (v4f*)(o + 20));
    __builtin_nontemporal_store(t6, (v4f*)(o + 24));
    __builtin_nontemporal_store(t7, (v4f*)(o + 28));
}

extern "C" void kernel_launch(void* const* d_in, const int* in_sizes, int n_in,
                              void* d_out, int out_size, void* d_ws, size_t ws_size,
                              hipStream_t stream) {
    const float* X = (const float*)d_in[0];
    float* out = (float*)d_out;
    const int npts = in_sizes[0] / 3;          // B*N*N points, 3 coords each
    const int blocks = (npts + PTS_PER_BLOCK - 1) / PTS_PER_BLOCK;
    spharm_all_kernel<<<blocks, PTS_PER_BLOCK, 0, stream>>>(X, out, npts);
}